// GCNForBiCls_57621281243476
// MI455X (gfx1250) — compile-verified
//
#include <hip/hip_runtime.h>

typedef __attribute__((ext_vector_type(2))) float v2f;
typedef __attribute__((ext_vector_type(8))) float v8f;

#define N_NODES 50000
#define N_EDGES 800000
#define N_FEAT  128
#define N_HID   64

// buf[n][f] = bias[f] — seeds the aggregation buffer so bias-add is free.
__global__ void init_bias_kernel(float* __restrict__ buf,
                                 const float* __restrict__ bias) {
    int i = blockIdx.x * blockDim.x + threadIdx.x;
    if (i < N_NODES * N_HID) buf[i] = bias[i & (N_HID - 1)];
}

// HW[M x 64] = X[M x K] @ W[K x 64] using V_WMMA_F32_16X16X4_F32.
// One wave per 16-row M-tile; 4 f32 accumulators cover all N=64.
// A-frag layout (f32 16x4): lanes 0-15 hold K=0,1; lanes 16-31 hold K=2,3.
// B-frag layout (f32 4x16): lanes 0-15 hold rows K=0,1; lanes 16-31 rows K=2,3.
__global__ __launch_bounds__(256) void gemm_wmma_kernel(
    const float* __restrict__ X, const float* __restrict__ W,
    float* __restrict__ HW, int M, int K) {
    int wave = (blockIdx.x * blockDim.x + threadIdx.x) >> 5;
    int lane = threadIdx.x & 31;
    int m0 = wave * 16;
    if (m0 >= M) return;                 // wave-uniform: EXEC stays all-ones
    int half = lane >> 4;                // 0 or 1
    int l    = lane & 15;

    v8f acc0 = {}, acc1 = {}, acc2 = {}, acc3 = {};
    const float* xrow = X + (size_t)(m0 + l) * K;

    for (int k = 0; k < K; k += 4) {
        int kk = k + 2 * half;
        v2f a = *(const v2f*)(xrow + kk);             // 8B-aligned pair load
        const float* bp = W + (size_t)kk * N_HID + l; // row kk, col base l
        v2f b0 = { bp[ 0], bp[ 0 + N_HID] };
        v2f b1 = { bp[16], bp[16 + N_HID] };
        v2f b2 = { bp[32], bp[32 + N_HID] };
        v2f b3 = { bp[48], bp[48 + N_HID] };
        acc0 = __builtin_amdgcn_wmma_f32_16x16x4_f32(false, a, false, b0, (short)0, acc0, false, false);
        acc1 = __builtin_amdgcn_wmma_f32_16x16x4_f32(false, a, false, b1, (short)0, acc1, false, false);
        acc2 = __builtin_amdgcn_wmma_f32_16x16x4_f32(false, a, false, b2, (short)0, acc2, false, false);
        acc3 = __builtin_amdgcn_wmma_f32_16x16x4_f32(false, a, false, b3, (short)0, acc3, false, false);
    }

    // C/D layout: VGPR r holds row m0+r (lanes 0-15) / m0+8+r (lanes 16-31), col = l.
    float* orow = HW + (size_t)(m0 + 8 * half) * N_HID + l;
    #pragma unroll
    for (int r = 0; r < 8; ++r) {
        orow[(size_t)r * N_HID +  0] = acc0[r];
        orow[(size_t)r * N_HID + 16] = acc1[r];
        orow[(size_t)r * N_HID + 32] = acc2[r];
        orow[(size_t)r * N_HID + 48] = acc3[r];
    }
}

// One thread per (edge, feature): agg[dst][f] += hw[src][f] * ew[e].
// Gather source and scatter target both fit in L2 (12.8 MB each).
__global__ void scatter_kernel(const float* __restrict__ HW,
                               const int* __restrict__ src,
                               const int* __restrict__ dst,
                               const float* __restrict__ ew,
                               float* __restrict__ agg) {
    int t = blockIdx.x * blockDim.x + threadIdx.x;
    if (t >= N_EDGES * N_HID) return;
    int e = t >> 6;          // edge
    int f = t & 63;          // feature
    float v = HW[(size_t)src[e] * N_HID + f] * ew[e];
    atomicAdd(&agg[(size_t)dst[e] * N_HID + f], v);
}

__global__ void relu_kernel(float* __restrict__ h) {
    int i = blockIdx.x * blockDim.x + threadIdx.x;
    if (i < N_NODES * N_HID) h[i] = fmaxf(h[i], 0.0f);
}

extern "C" void kernel_launch(void* const* d_in, const int* in_sizes, int n_in,
                              void* d_out, int out_size, void* d_ws, size_t ws_size,
                              hipStream_t stream) {
    const float* x  = (const float*)d_in[0];
    const int*   ei = (const int*)  d_in[1];   // [2, E]
    const float* ew = (const float*)d_in[2];
    const float* w1 = (const float*)d_in[3];
    const float* b1 = (const float*)d_in[4];
    const float* w2 = (const float*)d_in[5];
    const float* b2 = (const float*)d_in[6];
    float* out = (float*)d_out;

    float* hw  = (float*)d_ws;                      // [N_NODES, 64]
    float* agg = hw + (size_t)N_NODES * N_HID;      // [N_NODES, 64]

    const int* src = ei;
    const int* dst = ei + N_EDGES;

    const int nodeElems = N_NODES * N_HID;              // 3.2M
    const int edgeElems = N_EDGES * N_HID;              // 51.2M
    const int mTiles    = N_NODES / 16;                 // 3125 (exact)
    dim3 blk(256);
    dim3 gNode((nodeElems + 255) / 256);
    dim3 gEdge((edgeElems + 255) / 256);
    dim3 gGemm((mTiles + 7) / 8);                       // 8 waves per block

    // ---- Layer 1: h = relu(A @ (x @ w1) + b1) ----
    gemm_wmma_kernel<<<gGemm, blk, 0, stream>>>(x, w1, hw, N_NODES, N_FEAT);
    init_bias_kernel<<<gNode, blk, 0, stream>>>(agg, b1);
    scatter_kernel<<<gEdge, blk, 0, stream>>>(hw, src, dst, ew, agg);
    relu_kernel<<<gNode, blk, 0, stream>>>(agg);

    // ---- Layer 2: out = A @ (h @ w2) + b2 ----
    gemm_wmma_kernel<<<gGemm, blk, 0, stream>>>(agg, w2, hw, N_NODES, N_HID);
    init_bias_kernel<<<gNode, blk, 0, stream>>>(out, b2);
    scatter_kernel<<<gEdge, blk, 0, stream>>>(hw, src, dst, ew, out);
}